// TreeEncoder_16003048145658
// MI455X (gfx1250) — compile-verified
//
#include <hip/hip_runtime.h>

typedef __attribute__((ext_vector_type(2))) float v2f;
typedef __attribute__((ext_vector_type(8))) float v8f;

#define TREES  2048
#define KNODES 511
#define EDIM   128
#define DEPTH  9

// One workgroup per tree. 256 threads = 8 wave32 waves.
// LDS holds the full tree state: 511 x 128 f32 = 261,632 B (fits in 320KB WGP LDS).
__global__ __launch_bounds__(256, 1)
void tree_encoder_kernel(const float* __restrict__ emb,
                         const float* __restrict__ W,
                         const float* __restrict__ bias,
                         const int*  __restrict__ tokens,
                         float* __restrict__ out)
{
    __shared__ float hbuf[KNODES * EDIM];   // subtree-sum state per node
    __shared__ float pmax[256];             // partial max scratch

    const int tree = blockIdx.x;
    const int tid  = threadIdx.x;
    const int wave = tid >> 5;              // 0..7
    const int lane = tid & 31;
    const int mrow = lane & 15;             // M index (A) / N index (B,C,D)
    const int half = lane >> 4;             // selects K pair / D-row offset

    const int ni  = wave;                   // each wave owns one 16-dim column block
    const int col = ni * 16 + mrow;         // output dim for this lane's B/C/D

    // ---------- Phase 1: h = emb[tokens] @ W + b via V_WMMA_F32_16X16X4_F32 ----------
    // Hoist this wave's B fragments of W (constant across all 32 M-tiles).
    // f32 B 4x16 layout: lanes 0-15 -> K=0,1 ; lanes 16-31 -> K=2,3 ; N = lane&15.
    v2f bfrag[32];
#pragma unroll
    for (int kk = 0; kk < 32; ++kk) {
        const int k = kk * 4 + 2 * half;
        bfrag[kk].x = W[(k + 0) * EDIM + col];
        bfrag[kk].y = W[(k + 1) * EDIM + col];
    }
    const float bv   = bias[col];
    const int   base = tree * KNODES;

#pragma unroll 1
    for (int mi = 0; mi < 32; ++mi) {
        // f32 A 16x4 layout: lane supplies row M = lane&15, K pair 2*half.
        const int node   = mi * 16 + mrow;
        const int nclamp = node < KNODES ? node : KNODES - 1;   // pad row: addr-safe,
        const int token  = tokens[base + nclamp];               // its D row is never stored
        const float* arow = emb + (size_t)token * EDIM + 2 * half;

        v8f c = {};
#pragma unroll
        for (int kk = 0; kk < 32; ++kk) {
            v2f a = *(const v2f*)(arow + kk * 4);               // E[M][k+2h : k+2h+1]
            c = __builtin_amdgcn_wmma_f32_16x16x4_f32(
                    false, a, false, bfrag[kk], (short)0, c, false, false);
        }
        // D layout: c[r] = D[r + 8*half][col]
#pragma unroll
        for (int r = 0; r < 8; ++r) {
            const int orow = mi * 16 + r + 8 * half;
            if (orow < KNODES)
                hbuf[orow * EDIM + col] = c[r] + bv;
        }
    }
    __syncthreads();

    // ---------- Phase 2: bottom-up subtree accumulation (race-free: per-parent) ----------
    for (int l = DEPTH - 1; l >= 1; --l) {
        const int pFirst = (1 << (l - 1)) - 1;      // first parent (level l-1)
        const int total  = (1 << (l - 1)) * EDIM;
        for (int e = tid; e < total; e += 256) {
            const int p = pFirst + (e >> 7);
            const int d = e & (EDIM - 1);
            hbuf[p * EDIM + d] += hbuf[(2 * p + 1) * EDIM + d]
                                + hbuf[(2 * p + 2) * EDIM + d];
        }
        __syncthreads();
    }

    // ---------- Phase 3: per-tree elementwise max over nodes, clamp at 0 ----------
    {
        const int d    = tid & (EDIM - 1);
        const int part = tid >> 7;                  // 0: nodes [0,256), 1: [256,511)
        const int n0   = part * 256;
        const int n1   = part ? KNODES : 256;
        float m = -__builtin_inff();
        for (int n = n0; n < n1; ++n)
            m = fmaxf(m, hbuf[n * EDIM + d]);
        pmax[tid] = m;
    }
    __syncthreads();
    if (tid < EDIM) {
        const float m = fmaxf(pmax[tid], pmax[tid + EDIM]);
        out[tree * EDIM + tid] = fmaxf(m, 0.0f);
    }
}

extern "C" void kernel_launch(void* const* d_in, const int* in_sizes, int n_in,
                              void* d_out, int out_size, void* d_ws, size_t ws_size,
                              hipStream_t stream)
{
    // setup_inputs() order: emb_weight, W, b, tokens, parent, level, batch_id
    const float* emb    = (const float*)d_in[0];
    const float* W      = (const float*)d_in[1];
    const float* bias   = (const float*)d_in[2];
    const int*   tokens = (const int*)d_in[3];
    float*       out    = (float*)d_out;            // [2048, 128] f32

    tree_encoder_kernel<<<TREES, 256, 0, stream>>>(emb, W, bias, tokens, out);
}